// GATEncoder_42408507081343
// MI455X (gfx1250) — compile-verified
//
#include <hip/hip_runtime.h>
#include <math.h>

typedef float v2f __attribute__((ext_vector_type(2)));
typedef float v8f __attribute__((ext_vector_type(8)));

#define NNODES 50000
#define FDIM   128

// ---------------------------------------------------------------------------
// C[M,128] = A[M,128] @ B[128,128] using V_WMMA_F32_16X16X4_F32 (native f32).
// One wave -> one 16x16 output tile; 8 waves/block cover all 128 columns.
// ISA 7.12.2 layouts:
//   A 16x4 f32 : lanes 0-15 hold M=lane (v0=K0, v1=K1); lanes 16-31 K2/K3.
//   B 4x16 f32 : lanes 0-15 hold N=lane (v0=K0, v1=K1); lanes 16-31 K2/K3.
//   C/D 16x16  : VGPR j -> M=j (lanes 0-15), M=j+8 (lanes 16-31), N=lane&15.
// No divergence anywhere in this kernel: EXEC stays all-ones (WMMA requires it).
// ---------------------------------------------------------------------------
__global__ __launch_bounds__(256) void gemm_wmma_f32(const float* __restrict__ A,
                                                     const float* __restrict__ B,
                                                     float* __restrict__ C) {
  const int lane    = threadIdx.x & 31;
  const int wave    = threadIdx.x >> 5;
  const int rowBase = blockIdx.x * 16;
  const int colBase = wave * 16;
  const int r  = lane & 15;          // A row / B,C column within tile
  const int kb = (lane >> 4) * 2;    // half-wave K sub-offset (0 or 2)

  const float* __restrict__ arow = A + (size_t)(rowBase + r) * FDIM;

  v8f acc = {};
  for (int k = 0; k < FDIM; k += 4) {
    v2f a, b;
    a.x = arow[k + kb];
    a.y = arow[k + kb + 1];
    b.x = B[(size_t)(k + kb)     * FDIM + colBase + r];
    b.y = B[(size_t)(k + kb + 1) * FDIM + colBase + r];
    acc = __builtin_amdgcn_wmma_f32_16x16x4_f32(false, a, false, b,
                                                (short)0, acc, false, false);
  }

  const int mOff = (lane >> 4) * 8;
#pragma unroll
  for (int j = 0; j < 8; ++j) {
    C[(size_t)(rowBase + mOff + j) * FDIM + colBase + r] = acc[j];
  }
}

// ---------------------------------------------------------------------------
// Per-(node,head) attention coefficients: asrc/adst = <h[n,hd,:], att[hd,:]>
// ---------------------------------------------------------------------------
__global__ void alpha_kernel(const float* __restrict__ h,
                             const float* __restrict__ att_s,
                             const float* __restrict__ att_d,
                             float* __restrict__ asrc,
                             float* __restrict__ adst,
                             int heads, int chan, int total) {
  int tid = blockIdx.x * blockDim.x + threadIdx.x;
  if (tid >= total) return;
  int hd = tid % heads;
  int n  = tid / heads;
  const float* hp = h + (size_t)n * FDIM + hd * chan;
  const float* as = att_s + hd * chan;
  const float* ad = att_d + hd * chan;
  float ss = 0.f, sd = 0.f;
  for (int c = 0; c < chan; ++c) {
    float v = hp[c];
    ss += v * as[c];
    sd += v * ad[c];
  }
  asrc[tid] = ss;
  adst[tid] = sd;
}

__global__ void zero_f32(float* __restrict__ p, int n) {
  int tid = blockIdx.x * blockDim.x + threadIdx.x;
  if (tid < n) p[tid] = 0.0f;
}

// ---------------------------------------------------------------------------
// Edge pass A: e = leaky_relu(asrc[s]+adst[d]); ex = exp(e); den[d] += ex.
// (exp(e)/sum exp(e) == max-shifted softmax mathematically; e is O(1) here so
//  no overflow risk — saves an entire atomic-max pass.)
// Edges >= E_real are the implicit self loops (s = d = e - E_real).
// ---------------------------------------------------------------------------
__global__ void edge_softmax_num(const int* __restrict__ src,
                                 const int* __restrict__ dst,
                                 int e_real, int e_tot,
                                 const float* __restrict__ asrc,
                                 const float* __restrict__ adst,
                                 float* __restrict__ ex,
                                 float* __restrict__ den,
                                 int heads) {
  int tid = blockIdx.x * blockDim.x + threadIdx.x;
  int total = e_tot * heads;
  if (tid >= total) return;
  int e  = tid / heads;
  int hd = tid - e * heads;
  int s, d;
  if (e < e_real) { s = src[e]; d = dst[e]; }
  else            { s = e - e_real; d = s; }
  float v = asrc[s * heads + hd] + adst[d * heads + hd];
  v = (v > 0.0f) ? v : 0.2f * v;          // leaky_relu, PyG default slope
  float w = expf(v);
  ex[tid] = w;
  atomicAdd(&den[d * heads + hd], w);
}

// ---------------------------------------------------------------------------
// Edge pass B: out[d, c] += (ex/den) * h[s, c]. 128 lanes per edge, 2 edges
// per 256-thread block. Atomics resolve in L2 (out fits in 192 MB L2).
// ---------------------------------------------------------------------------
__global__ __launch_bounds__(256) void edge_aggregate(const int* __restrict__ src,
                                                      const int* __restrict__ dst,
                                                      int e_real, int e_tot,
                                                      const float* __restrict__ h,
                                                      const float* __restrict__ ex,
                                                      const float* __restrict__ den,
                                                      float* __restrict__ out,
                                                      int heads, int chanShift) {
  int e = blockIdx.x * 2 + (threadIdx.x >> 7);
  if (e >= e_tot) return;
  int c  = threadIdx.x & 127;
  int hd = c >> chanShift;               // chan = 32 (shift 5) or 128 (shift 7)
  int s, d;
  if (e < e_real) { s = src[e]; d = dst[e]; }
  else            { s = e - e_real; d = s; }
  float w = ex[e * heads + hd] / den[d * heads + hd];
  atomicAdd(&out[(size_t)d * FDIM + c], w * h[(size_t)s * FDIM + c]);
}

__global__ void bias_elu(float* __restrict__ x, const float* __restrict__ b, int total) {
  int tid = blockIdx.x * blockDim.x + threadIdx.x;
  if (tid >= total) return;
  float v = x[tid] + b[tid & 127];
  x[tid] = (v > 0.0f) ? v : (expf(v) - 1.0f);
}

__global__ void bias_out(const float* __restrict__ x, const float* __restrict__ b,
                         float* __restrict__ out, int total) {
  int tid = blockIdx.x * blockDim.x + threadIdx.x;
  if (tid >= total) return;
  out[tid] = x[tid] + b[tid & 127];
}

// ---------------------------------------------------------------------------
extern "C" void kernel_launch(void* const* d_in, const int* in_sizes, int n_in,
                              void* d_out, int out_size, void* d_ws, size_t ws_size,
                              hipStream_t stream) {
  const float* x   = (const float*)d_in[0];
  const int*   ei  = (const int*)  d_in[1];
  const float* W1  = (const float*)d_in[2];
  const float* as1 = (const float*)d_in[3];
  const float* ad1 = (const float*)d_in[4];
  const float* b1  = (const float*)d_in[5];
  const float* W2  = (const float*)d_in[6];
  const float* as2 = (const float*)d_in[7];
  const float* ad2 = (const float*)d_in[8];
  const float* b2  = (const float*)d_in[9];
  float* out = (float*)d_out;

  const int N    = NNODES;
  const int E    = in_sizes[1] / 2;      // 800000
  const int Etot = E + N;                // + self loops
  const int* srcI = ei;
  const int* dstI = ei + E;

  // workspace layout (floats): ~67.2 MB total
  float* ws     = (float*)d_ws;
  float* bufH   = ws;                                 // [N,128] pre-attention h
  float* bufAgg = bufH   + (size_t)N * FDIM;          // [N,128] attention output
  float* asrc   = bufAgg + (size_t)N * FDIM;          // [N,heads]
  float* adst   = asrc   + (size_t)N * 4;
  float* den    = adst   + (size_t)N * 4;
  float* ex     = den    + (size_t)N * 4;             // [Etot,heads]

  const int NT = 256;
  auto cdiv = [](int a, int b) { return (a + b - 1) / b; };

  // ================= layer 1: heads=4, out_ch=32, concat =================
  gemm_wmma_f32<<<N / 16, NT, 0, stream>>>(x, W1, bufH);
  alpha_kernel<<<cdiv(N * 4, NT), NT, 0, stream>>>(bufH, as1, ad1, asrc, adst, 4, 32, N * 4);
  zero_f32<<<cdiv(N * 4, NT), NT, 0, stream>>>(den, N * 4);
  zero_f32<<<cdiv(N * FDIM, NT), NT, 0, stream>>>(bufAgg, N * FDIM);
  edge_softmax_num<<<cdiv(Etot * 4, NT), NT, 0, stream>>>(srcI, dstI, E, Etot,
                                                          asrc, adst, ex, den, 4);
  edge_aggregate<<<cdiv(Etot, 2), NT, 0, stream>>>(srcI, dstI, E, Etot,
                                                   bufH, ex, den, bufAgg, 4, 5);
  bias_elu<<<cdiv(N * FDIM, NT), NT, 0, stream>>>(bufAgg, b1, N * FDIM);

  // ================= layer 2: heads=1, out_ch=128, mean ===================
  gemm_wmma_f32<<<N / 16, NT, 0, stream>>>(bufAgg, W2, bufH);
  alpha_kernel<<<cdiv(N, NT), NT, 0, stream>>>(bufH, as2, ad2, asrc, adst, 1, 128, N);
  zero_f32<<<cdiv(N, NT), NT, 0, stream>>>(den, N);
  zero_f32<<<cdiv(N * FDIM, NT), NT, 0, stream>>>(bufAgg, N * FDIM);  // safe: gemm done first
  edge_softmax_num<<<cdiv(Etot, NT), NT, 0, stream>>>(srcI, dstI, E, Etot,
                                                      asrc, adst, ex, den, 1);
  edge_aggregate<<<cdiv(Etot, 2), NT, 0, stream>>>(srcI, dstI, E, Etot,
                                                   bufH, ex, den, bufAgg, 1, 7);
  bias_out<<<cdiv(N * FDIM, NT), NT, 0, stream>>>(bufAgg, b2, out, N * FDIM);
}